// CRX_88871463288929
// MI455X (gfx1250) — compile-verified
//
#include <hip/hip_runtime.h>

// Controlled-RX on 12 qubits, D=4096 basis states, B=128 complex64 state vectors.
//
// Index algebra of the reference (C=0 control bit = bit 11, T=1 target bit = bit 10):
//   rows [0,2048)    : identity  -> streaming float4 copy (2 MB in + 2 MB out)
//   rows [2048,4096) : pairs (ij, ij+1024), ij in [2048,3072), rotated by the
//                      4x4 real matrix  M = [ c 0 0  s;  0 c -s 0;  0 s c 0; -s 0 0 c ]
//                      acting on (re_ij, im_ij, re_ik, im_ik), c=cos(a/2), s=sin(a/2)
//
// Roofline: 8 MB total traffic @ 23.3 TB/s => ~0.35 us; compute ~5 MFLOP (negligible).
// Pure bandwidth problem. The rotated half is expressed as 4 chained
// V_WMMA_F32_16X16X4_F32 K-slices, one pair per slice (M in A-rows 4k..4k+3).
// The interleaved (re,im) complex64 layout coincides exactly with the B-matrix
// VGPR layout (lane=column, K0/K1 = re/im of row ij in lanes 0-15, K2/K3 =
// re/im of row ik in lanes 16-31): one b64 load per lane per slice, no LDS,
// no shuffles. Accumulator layout gives coalesced b64 (float2) stores.
//
// A-matrix entries are built branchlessly:  M[lr][K] = c  iff lr==K,
// ±s iff lr+K==3 (+ for even lr, - for odd), zero otherwise — both conditions
// self-vanish for lr outside [0,4), so no EXEC-divergent guards and the four
// global loads can all be issued before the WMMA chain.

typedef float v2f __attribute__((ext_vector_type(2)));
typedef float v8f __attribute__((ext_vector_type(8)));

#define NCOLS 128
#define COPY_BLOCKS 512   // 512*256 threads * 16B = 2 MB  (rows [0,2048))
#define WMMA_BLOCKS 256   // 256 blocks * 8 waves = 2048 wave-tiles

__global__ __launch_bounds__(256) void crx_wmma_kernel(
    const float2* __restrict__ x,     // complex64 as (re,im) float2, row-major (D,B)
    const float*  __restrict__ angle, // single element
    float2*       __restrict__ out)
{
    if ((int)blockIdx.x < COPY_BLOCKS) {
        // ---- identity half: rows [0,2048), bulk float4 streaming copy ----
        const float4* __restrict__ src = (const float4*)x;
        float4* __restrict__ dst = (float4*)out;
        int u = (int)blockIdx.x * 256 + (int)threadIdx.x;   // [0, 131072)
        dst[u] = src[u];
        return;
    }

    // ---- rotated half: rows [2048,4096) via WMMA f32 16x16x4 ----
    const int wave = ((int)blockIdx.x - COPY_BLOCKS) * 8 + ((int)threadIdx.x >> 5); // [0,2048)
    const int lane = (int)threadIdx.x & 31;
    const int gr   = wave >> 3;          // pair-group: pairs 4*gr .. 4*gr+3
    const int t    = wave & 7;           // column tile: cols 16*t .. 16*t+15
    const int n0   = lane & 15;          // A row / B,D column within tile
    const int h    = lane >> 4;          // 0: K0/K1 half, 1: K2/K3 half
    const int col  = t * 16 + n0;

    // Load row base: h=0 -> rows 2048+p (ij), h=1 -> rows 3072+p (ik), p = 4*gr+k.
    // Per-k offset is +128 float2 => folds into the instruction offset.
    const int ldbase = (2048 + 1024 * h + 4 * gr) * NCOLS + col;

    // Issue all four b64 loads up front (lands directly in B-matrix layout).
    float2 d0 = x[ldbase + 0 * NCOLS];
    float2 d1 = x[ldbase + 1 * NCOLS];
    float2 d2 = x[ldbase + 2 * NCOLS];
    float2 d3 = x[ldbase + 3 * NCOLS];

    float s, c;
    __sincosf(angle[0] * 0.5f, &s, &c);
    const float sv = (n0 & 1) ? -s : s;  // sign of the s-entry depends on row parity
    const int  k0  = 2 * h;              // K index held in A VGPR0
    const int  k1  = k0 + 1;             // K index held in A VGPR1

    v8f acc = (v8f){0.f, 0.f, 0.f, 0.f, 0.f, 0.f, 0.f, 0.f};

#pragma unroll
    for (int k = 0; k < 4; ++k) {
        // Branchless A: lr = row within pair-k's 4x4 block (garbage outside,
        // but both selects are then false => contributes exact zeros).
        const int lr = n0 - 4 * k;
        v2f A;
        A[0] = ((lr == k0) ? c : 0.f) + ((lr + k0 == 3) ? sv : 0.f);
        A[1] = ((lr == k1) ? c : 0.f) + ((lr + k1 == 3) ? sv : 0.f);

        const float2 d = (k == 0) ? d0 : (k == 1) ? d1 : (k == 2) ? d2 : d3;
        v2f B; B[0] = d.x; B[1] = d.y;

        acc = __builtin_amdgcn_wmma_f32_16x16x4_f32(
            /*neg_a=*/false, A, /*neg_b=*/false, B,
            /*c_mod=*/(short)0, acc, /*reuse_a=*/false, /*reuse_b=*/false);
    }

    // D layout: VGPR g = row g (lanes 0-15) / row g+8 (lanes 16-31), col = n0.
    // Row 4*kp+comp = pair (4*gr + kp + 2*h), comp in {re_ij, im_ij, re_ik, im_ik}.
    // Adjacent accumulator VGPRs give (re,im) -> coalesced b64 stores.
    const int stbase = (2048 + 4 * gr + 2 * h) * NCOLS + col;
#pragma unroll
    for (int kp = 0; kp < 2; ++kp) {
        float2 oij; oij.x = acc[4 * kp + 0]; oij.y = acc[4 * kp + 1];
        float2 oik; oik.x = acc[4 * kp + 2]; oik.y = acc[4 * kp + 3];
        out[stbase + kp * NCOLS] = oij;                  // row ij
        out[stbase + kp * NCOLS + 1024 * NCOLS] = oik;   // row ik
    }
}

extern "C" void kernel_launch(void* const* d_in, const int* in_sizes, int n_in,
                              void* d_out, int out_size, void* d_ws, size_t ws_size,
                              hipStream_t stream) {
    (void)in_sizes; (void)n_in; (void)out_size; (void)d_ws; (void)ws_size;
    const float2* x     = (const float2*)d_in[0];   // complex64 -> float2
    const float*  angle = (const float*)d_in[1];    // 1-element
    float2*       out   = (float2*)d_out;

    dim3 grid(COPY_BLOCKS + WMMA_BLOCKS);
    dim3 block(256);
    hipLaunchKernelGGL(crx_wmma_kernel, grid, block, 0, stream, x, angle, out);
}